// GCNLayer_33535104647603
// MI455X (gfx1250) — compile-verified
//
#include <hip/hip_runtime.h>
#include <hip/hip_bf16.h>

// MI455X (gfx1250) GCN layer:
//   L1 = spmm(fea) + b0 ; L2 = spmm(L1) + b1 ; out = (fea + L1 + L2)/3
// L2-cache-resident workload (everything < 192MB). Edge-parallel SpMM with
// wave32-per-edge float4 gathers + global f32 atomic scatter; bias broadcasts
// done as rank-1 WMMA updates (v_wmma_f32_16x16x4_f32) on 16x16 tiles.

#define HIDDEN 128

typedef float v2f __attribute__((ext_vector_type(2)));
typedef float v8f __attribute__((ext_vector_type(8)));

// ---------------------------------------------------------------------------
// Kernel 1: zero the two accumulators in workspace (ws is poisoned 0xAA).
// ---------------------------------------------------------------------------
__global__ void gcn_zero_ws(float4* __restrict__ p, unsigned long long n4) {
    unsigned long long i = (unsigned long long)blockIdx.x * blockDim.x + threadIdx.x;
    unsigned long long stride = (unsigned long long)gridDim.x * blockDim.x;
    float4 z = make_float4(0.f, 0.f, 0.f, 0.f);
    for (; i < n4; i += stride) p[i] = z;
}

// ---------------------------------------------------------------------------
// Kernel 2/4: edge-parallel SpMM.  One wave32 per edge:
//   lane l handles features [4l, 4l+4): one coalesced float4 gather of the
//   source row (L2 hit: feature matrix is 25.6MB), scale by val, and four
//   global_atomic_add_f32 scatters to the destination row.
// Edge index e is wave-uniform -> row/col/val become scalar (SMEM) loads.
// ---------------------------------------------------------------------------
__global__ void gcn_spmm_edges(const float* __restrict__ src,
                               const int*   __restrict__ row,
                               const int*   __restrict__ col,
                               const float* __restrict__ val,
                               float*       __restrict__ dst,
                               int n_edges) {
    const int lane   = threadIdx.x & 31;
    const int wave   = threadIdx.x >> 5;
    const int nwaves = (gridDim.x * blockDim.x) >> 5;
    int e = blockIdx.x * (blockDim.x >> 5) + wave;

    for (; e < n_edges; e += nwaves) {
        // gfx1250 speculative prefetch of the edge streams (global_prefetch_b8)
        __builtin_prefetch(col + e + 4096, 0, 0);
        __builtin_prefetch(row + e + 4096, 0, 0);

        const int   r = row[e];
        const int   c = col[e];
        const float v = val[e];

        const float4* s4 = (const float4*)(src + (size_t)c * HIDDEN);
        float4 x = s4[lane];

        float* d = dst + (size_t)r * HIDDEN + lane * 4;
        atomicAdd(d + 0, v * x.x);
        atomicAdd(d + 1, v * x.y);
        atomicAdd(d + 2, v * x.z);
        atomicAdd(d + 3, v * x.w);
    }
}

// ---------------------------------------------------------------------------
// Rank-1 bias broadcast via WMMA:  D = A(16x4) * B(4x16) + C(16x16)
//   A rows = [1,0,0,0]  (lanes 0-15 hold K=0/K=1, lanes 16-31 hold K=2/K=3)
//   B row0 = bias chunk (lanes 0-15 hold K=0/K=1, lanes 16-31 hold K=2/K=3)
// => D[m,n] = bias[n] + C[m,n] exactly (1.0*b + 0*x, RNE).
// ---------------------------------------------------------------------------
__device__ __forceinline__ v8f wmma_bias_add(v8f c, float bias_lane, int lane) {
    v2f a, b;
    a.x = (lane < 16) ? 1.0f : 0.0f;  // K=0 column is ones; K=2 column is zero
    a.y = 0.0f;                       // K=1 / K=3 columns are zero
    b.x = bias_lane;                  // K=0 row = bias (K=2 row killed by A=0)
    b.y = 0.0f;                       // keep finite: 0 * Inf/NaN would poison
    return __builtin_amdgcn_wmma_f32_16x16x4_f32(
        /*neg_a=*/false, a, /*neg_b=*/false, b,
        /*c_mod=*/(short)0, c, /*reuse_a=*/false, /*reuse_b=*/false);
}

// C/D 16x16 f32 tile layout: VGPR k -> row (lane<16 ? row0+k : row0+8+k),
// column = col0 + (lane & 15).  One wave per tile; EXEC all-1s.

// Kernel 3: x[tile] += bias  (in place; x becomes L1)
__global__ void gcn_bias_wmma(float* __restrict__ x,
                              const float* __restrict__ bias,
                              int n_tiles) {
    const int lane = threadIdx.x & 31;
    const int t    = blockIdx.x * 8 + (threadIdx.x >> 5);  // 256 thr = 8 waves
    if (t >= n_tiles) return;                              // whole-wave exit

    const int row0 = (t >> 3) << 4;            // 8 column tiles (HIDDEN/16)
    const int col0 = (t & 7) << 4;
    const int r    = row0 + ((lane >> 4) << 3); // +8 for upper half-wave
    const int cix  = col0 + (lane & 15);

    float* p = x + (size_t)r * HIDDEN + cix;
    v8f c;
#pragma unroll
    for (int k = 0; k < 8; ++k) c[k] = p[k * HIDDEN];

    v8f d = wmma_bias_add(c, bias[cix - row0 * 0 - col0 + col0] /* = bias[cix] over 0..127? no: */, lane);
    // NOTE: bias index must be the hidden-dim position = cix (col0 + lane&15)
#pragma unroll
    for (int k = 0; k < 8; ++k) p[k * HIDDEN] = d[k];
}

// Kernel 5: out = (fea + L1 + acc2 + 1*b1^T) / 3
__global__ void gcn_final_wmma(const float* __restrict__ fea,
                               const float* __restrict__ l1,
                               const float* __restrict__ acc2,
                               const float* __restrict__ bias1,
                               float* __restrict__ out,
                               int n_tiles) {
    const int lane = threadIdx.x & 31;
    const int t    = blockIdx.x * 8 + (threadIdx.x >> 5);
    if (t >= n_tiles) return;

    const int row0 = (t >> 3) << 4;
    const int col0 = (t & 7) << 4;
    const int r    = row0 + ((lane >> 4) << 3);
    const int cix  = col0 + (lane & 15);
    const size_t base = (size_t)r * HIDDEN + cix;

    v8f c;
#pragma unroll
    for (int k = 0; k < 8; ++k) {
        size_t idx = base + (size_t)k * HIDDEN;
        c[k] = fea[idx] + l1[idx] + acc2[idx];
    }

    v8f d = wmma_bias_add(c, bias1[cix], lane);  // + b1 broadcast

    const float third = 1.0f / 3.0f;
    float* q = out + base;
#pragma unroll
    for (int k = 0; k < 8; ++k) q[k * HIDDEN] = d[k] * third;
}

// ---------------------------------------------------------------------------
extern "C" void kernel_launch(void* const* d_in, const int* in_sizes, int n_in,
                              void* d_out, int out_size, void* d_ws, size_t ws_size,
                              hipStream_t stream) {
    const float* fea  = (const float*)d_in[0];
    const int*   row  = (const int*)  d_in[1];
    const int*   col  = (const int*)  d_in[2];
    const float* val  = (const float*)d_in[3];
    const float* bias = (const float*)d_in[4];   // [2, 128]
    float*       out  = (float*)d_out;

    const int n_nodes = in_sizes[0] / HIDDEN;    // 50000
    const int n_edges = in_sizes[1];             // 500000

    float* acc1 = (float*)d_ws;                              // 25.6 MB
    float* acc2 = acc1 + (size_t)n_nodes * HIDDEN;           // 25.6 MB

    // 1) zero both accumulators
    {
        unsigned long long n4 = (unsigned long long)2 * n_nodes * HIDDEN / 4;
        int blocks = (int)((n4 + 255) / 256);
        if (blocks > 16384) blocks = 16384;
        gcn_zero_ws<<<blocks, 256, 0, stream>>>((float4*)acc1, n4);
    }

    const int spmm_blocks = (n_edges + 7) / 8;   // 8 waves (edges) per block
    const int n_tiles     = (n_nodes / 16) * (HIDDEN / 16);
    const int tile_blocks = (n_tiles + 7) / 8;

    // 2) acc1 = A * fea
    gcn_spmm_edges<<<spmm_blocks, 256, 0, stream>>>(fea, row, col, val, acc1, n_edges);
    // 3) acc1 += b0   (acc1 becomes L1)
    gcn_bias_wmma<<<tile_blocks, 256, 0, stream>>>(acc1, bias, n_tiles);
    // 4) acc2 = A * L1
    gcn_spmm_edges<<<spmm_blocks, 256, 0, stream>>>(acc1, row, col, val, acc2, n_edges);
    // 5) out = (fea + L1 + acc2 + b1) / 3
    gcn_final_wmma<<<tile_blocks, 256, 0, stream>>>(fea, acc1, acc2, bias + HIDDEN,
                                                    out, n_tiles);
}